// MeshLoss_53867479827007
// MI455X (gfx1250) — compile-verified
//
#include <hip/hip_runtime.h>
#include <hip/hip_bf16.h>
#include <stdint.h>

// ---------------- CDNA5 async global->LDS copy plumbing ----------------
#ifndef __has_builtin
#define __has_builtin(x) 0
#endif

#if __has_builtin(__builtin_amdgcn_global_load_async_to_lds_b128)
#define ASYNC_BUILTIN 1
#else
#define ASYNC_BUILTIN 0
#endif

typedef int v4i_t __attribute__((ext_vector_type(4)));
typedef __attribute__((address_space(1))) v4i_t gv4i_t;  // global
typedef __attribute__((address_space(3))) v4i_t lv4i_t;  // LDS

// explicit-LDS pointer types for the compute side (avoid flat-path reads)
typedef const __attribute__((address_space(3))) char*  lcc_t;
typedef const __attribute__((address_space(3))) float* lcf_t;

// One b128 async transfer at compile-time byte offset OFF.
// (INST_OFFSET applies to BOTH addresses -> share one base VGPR set per tile)
template <int OFF>
__device__ __forceinline__ void async_cp16_off(const void* g, void* l) {
#if ASYNC_BUILTIN
  __builtin_amdgcn_global_load_async_to_lds_b128(
      (gv4i_t*)(void*)(uintptr_t)g, (lv4i_t*)l, /*offset=*/OFF, /*cpol=*/0);
#else
  unsigned loff = (unsigned)(uintptr_t)(__attribute__((address_space(3))) void*)l;
  unsigned long long ga = (unsigned long long)(uintptr_t)g;
  asm volatile("global_load_async_to_lds_b128 %0, %1, off offset:%2"
               :: "v"(loff), "v"(ga), "i"(OFF) : "memory");
#endif
}

// compile-time unroll so OFF stays an integer constant expression
template <int I, int NTOT>
struct Issuer {
  static __device__ __forceinline__ void run(const void* g, void* l) {
    async_cp16_off<I * 512>(g, l);
    Issuer<I + 1, NTOT>::run(g, l);
  }
};
template <int NTOT>
struct Issuer<NTOT, NTOT> {
  static __device__ __forceinline__ void run(const void*, void*) {}
};

#if __has_builtin(__builtin_amdgcn_s_wait_asynccnt)
#define WAIT_ASYNC(n) do { asm volatile("" ::: "memory");            \
                           __builtin_amdgcn_s_wait_asynccnt(n);      \
                           asm volatile("" ::: "memory"); } while (0)
#else
#define WAIT_ASYNC(n) asm volatile("s_wait_asynccnt %0" :: "i"(n) : "memory")
#endif

// ---------------- geometry of the staged tiles ----------------
#define WPB      2          // waves per block
#define BLOCK    64         // threads per block (wave32)
#define TILE     128        // tets per wave-tile
#define INV_TB   4608       // 128 * 36 B  = 9  * 512 B  (b128 x 9  per wave)
#define MESH_TB  6144       // 128 * 48 B  = 12 * 512 B  (b128 x 12 per wave)
#define TILE_B   (INV_TB + MESH_TB)   // 10752 B, 16B-aligned
#define NISSUE   21         // async instructions per tile (9 + 12)

// ---------------- per-tet closed-form loss (pointer-type generic) ----------------
template <typename P>
__device__ __forceinline__ float tet_loss_t(P iv, P m) {
  // diff[j][k] = mesh_out[1+j][k] - mesh_out[0][k]
  float d00 = m[3] - m[0],  d01 = m[4]  - m[1],  d02 = m[5]  - m[2];
  float d10 = m[6] - m[0],  d11 = m[7]  - m[1],  d12 = m[8]  - m[2];
  float d20 = m[9] - m[0],  d21 = m[10] - m[1],  d22 = m[11] - m[2];
  // A = inv @ diff
  float a00 = iv[0]*d00 + iv[1]*d10 + iv[2]*d20;
  float a01 = iv[0]*d01 + iv[1]*d11 + iv[2]*d21;
  float a02 = iv[0]*d02 + iv[1]*d12 + iv[2]*d22;
  float a10 = iv[3]*d00 + iv[4]*d10 + iv[5]*d20;
  float a11 = iv[3]*d01 + iv[4]*d11 + iv[5]*d21;
  float a12 = iv[3]*d02 + iv[4]*d12 + iv[5]*d22;
  float a20 = iv[6]*d00 + iv[7]*d10 + iv[8]*d20;
  float a21 = iv[6]*d01 + iv[7]*d11 + iv[8]*d21;
  float a22 = iv[6]*d02 + iv[7]*d12 + iv[8]*d22;
  // G = A^T A (symmetric); loss = ||G - I||_F^2
  float g00 = a00*a00 + a10*a10 + a20*a20;
  float g11 = a01*a01 + a11*a11 + a21*a21;
  float g22 = a02*a02 + a12*a12 + a22*a22;
  float g01 = a00*a01 + a10*a11 + a20*a21;
  float g02 = a00*a02 + a10*a12 + a20*a22;
  float g12 = a01*a02 + a11*a12 + a21*a22;
  float t0 = g00 - 1.0f, t1 = g11 - 1.0f, t2 = g22 - 1.0f;
  return t0*t0 + t1*t1 + t2*t2 + 2.0f*(g01*g01 + g02*g02 + g12*g12);
}

// issue one wave-tile: 9 + 12 async b128 transfers (512 B per wave-instr),
// all sharing one base address per array via the offset immediate
__device__ __forceinline__ void issue_tile(const char* gi, const char* gm,
                                           char* l, int lane) {
  const int lo = lane * 16;
  Issuer<0, 9>::run(gi + lo, l + lo);
  Issuer<0, 12>::run(gm + lo, l + INV_TB + lo);
}

__global__ void __launch_bounds__(BLOCK)
mesh_loss_partial(const float* __restrict__ inv,
                  const float* __restrict__ mesh,
                  float* __restrict__ partials, int N) {
  __shared__ __align__(16) char smem[2 * WPB * TILE_B];  // double-buffered per wave
  __shared__ float red[BLOCK];

  const int tid  = threadIdx.x;
  const int lane = tid & 31;
  const int w    = tid >> 5;
  float acc = 0.0f;

  const int numTiles = N / TILE;
  const int gw      = blockIdx.x * WPB + w;       // global wave id
  const int wstride = gridDim.x * WPB;

  char* lbuf[2];
  lbuf[0] = smem + (0 * WPB + w) * TILE_B;
  lbuf[1] = smem + (1 * WPB + w) * TILE_B;
  const char* ginv  = (const char*)inv;
  const char* gmesh = (const char*)mesh;

  long t = gw;
  int buf = 0;
  if (t < numTiles)
    issue_tile(ginv + t * (long)INV_TB, gmesh + t * (long)MESH_TB, lbuf[0], lane);

  for (; t < numTiles; t += wstride) {
    const long tn  = t + wstride;
    const bool more = tn < numTiles;
    char* cur = lbuf[buf];
    if (more)  // prefetch next tile into the other buffer before waiting
      issue_tile(ginv + tn * (long)INV_TB, gmesh + tn * (long)MESH_TB,
                 lbuf[buf ^ 1], lane);
    if (more) { WAIT_ASYNC(NISSUE); }  // loads complete in-order: current tile done
    else      { WAIT_ASYNC(0); }

    // explicit AS3 pointers -> ds_load path (not flat)
    lcc_t li = (lcc_t)cur;
    lcc_t lm = li + INV_TB;
#pragma unroll
    for (int s = 0; s < TILE / 32; ++s) {
      const int e = s * 32 + lane;
      acc += tet_loss_t<lcf_t>((lcf_t)(li + e * 36), (lcf_t)(lm + e * 48));
    }
    buf ^= 1;
  }

  // tail (N % 128): direct global loads, block 0 only — deterministic
  const long tailStart = (long)numTiles * TILE;
  if (blockIdx.x == 0) {
    for (long i = tailStart + tid; i < N; i += BLOCK)
      acc += tet_loss_t<const float*>(inv + 9 * i, mesh + 12 * i);
  }

  // deterministic in-block reduction (ordered sum by thread 0)
  red[tid] = acc;
  __syncthreads();
  if (tid == 0) {
    float s = 0.0f;
#pragma unroll 8
    for (int i = 0; i < BLOCK; ++i) s += red[i];
    partials[blockIdx.x] = s;
  }
}

__global__ void reduce_partials(const float* __restrict__ p, int n,
                                float* __restrict__ out, float scale) {
  __shared__ float sh[256];
  float a = 0.0f;
  for (int i = threadIdx.x; i < n; i += 256) a += p[i];  // fixed stride: deterministic
  sh[threadIdx.x] = a;
  __syncthreads();
  if (threadIdx.x == 0) {
    float t = 0.0f;
    for (int i = 0; i < 256; ++i) t += sh[i];
    out[0] = t * scale;
  }
}

extern "C" void kernel_launch(void* const* d_in, const int* in_sizes, int n_in,
                              void* d_out, int out_size, void* d_ws, size_t ws_size,
                              hipStream_t stream) {
  const float* inv  = (const float*)d_in[0];   // [N,3,3] f32
  const float* mesh = (const float*)d_in[1];   // [N,4,3] f32
  float* out      = (float*)d_out;
  float* partials = (float*)d_ws;

  const int N = in_sizes[0] / 9;

  int numBlocks = 2048;                         // 4096 waves; ~3.8 tiles/wave at N=2M
  const int maxp = (int)(ws_size / sizeof(float));
  if (maxp > 0 && numBlocks > maxp) numBlocks = maxp;
  if (numBlocks < 1) numBlocks = 1;

  mesh_loss_partial<<<numBlocks, BLOCK, 0, stream>>>(inv, mesh, partials, N);
  const float scale = (N > 0) ? (1.0f / (float)N) : 0.0f;
  reduce_partials<<<1, 256, 0, stream>>>(partials, numBlocks, out, scale);
}